// LANTM_5866925326740
// MI455X (gfx1250) — compile-verified
//
#include <hip/hip_runtime.h>
#include <math.h>

#define B_   512
#define N_   1024
#define M_   64
#define H_   512
#define IN_  256
#define NH_  4
#define EPS_ 1e-8f

typedef float v2f __attribute__((ext_vector_type(2)));
typedef float v8f __attribute__((ext_vector_type(8)));

__device__ __forceinline__ float sigm(float x) { return 1.f / (1.f + expf(-x)); }
__device__ __forceinline__ float softplus(float x) {
    return (x > 20.f) ? x : log1pf(expf(x));
}

__device__ __forceinline__ v8f wmma4(v2f a, v2f b, v8f c) {
    return __builtin_amdgcn_wmma_f32_16x16x4_f32(
        false, a, false, b, (short)0, c, false, false);
}

// ---------------------------------------------------------------------------
// f32 WMMA GEMM, 32x32 tile per wave (register-blocked):
//   C[rows x J] = A[rows x K] @ W[J x K]^T + bias
// 2 A-frags x 2 B-frags -> 4 V_WMMA_F32_16X16X4_F32 per K-step of 4, giving
// 4 independent accumulation chains and 8 B/lane of VMEM per WMMA.
// Fragment layout per ISA: float2 at (row = base + lane%16, col = k+2*(lane>>4));
// D: vgpr r -> row r + 8*(lane>>4), col = lane%16.
// ACT: 0 none, 1 sigmoid.  ACCUM: 1 -> C += result.
// K must be a multiple of 16.
// ---------------------------------------------------------------------------
template <int ACT, int ACCUM>
__global__ void wmma_gemm_f32(const float* __restrict__ A,
                              const float* __restrict__ W,
                              const float* __restrict__ bias,
                              float* __restrict__ C,
                              int rows, int K, int J) {
    const int lane  = threadIdx.x & 31;
    const int wave  = blockIdx.x * (blockDim.x >> 5) + (threadIdx.x >> 5);
    const int tilesJ = (J + 31) >> 5;
    const int tilesB = (rows + 31) >> 5;
    if (wave >= tilesJ * tilesB) return;     // wave-uniform exit (EXEC stays all-1)

    const int row0 = (wave / tilesJ) << 5;
    const int col0 = (wave % tilesJ) << 5;
    const int lr   = lane & 15;
    const int hi   = lane >> 4;

    int ar0 = row0 + lr;       if (ar0 >= rows) ar0 = rows - 1;
    int ar1 = row0 + 16 + lr;  if (ar1 >= rows) ar1 = rows - 1;
    int wr0 = col0 + lr;       if (wr0 >= J)    wr0 = J - 1;
    int wr1 = col0 + 16 + lr;  if (wr1 >= J)    wr1 = J - 1;
    const float* Ap0 = A + (size_t)ar0 * K + 2 * hi;
    const float* Ap1 = A + (size_t)ar1 * K + 2 * hi;
    const float* Wp0 = W + (size_t)wr0 * K + 2 * hi;
    const float* Wp1 = W + (size_t)wr1 * K + 2 * hi;

    v8f acc00 = {0.f,0.f,0.f,0.f,0.f,0.f,0.f,0.f};
    v8f acc01 = acc00, acc10 = acc00, acc11 = acc00;

    for (int k = 0; k < K; k += 16) {
        // pull next 256B chunk of each stream toward the near cache
        // (speculative prefetch: OOB at tile end is silently dropped)
        __builtin_prefetch(Ap0 + k + 64, 0, 3);
        __builtin_prefetch(Ap1 + k + 64, 0, 3);
        __builtin_prefetch(Wp0 + k + 64, 0, 3);
        __builtin_prefetch(Wp1 + k + 64, 0, 3);
#pragma unroll
        for (int kk = 0; kk < 16; kk += 4) {
            v2f a0 = *(const v2f*)(Ap0 + k + kk);
            v2f a1 = *(const v2f*)(Ap1 + k + kk);
            v2f b0 = *(const v2f*)(Wp0 + k + kk);
            v2f b1 = *(const v2f*)(Wp1 + k + kk);
            acc00 = wmma4(a0, b0, acc00);
            acc01 = wmma4(a0, b1, acc01);
            acc10 = wmma4(a1, b0, acc10);
            acc11 = wmma4(a1, b1, acc11);
        }
    }

    float bv0 = 0.f, bv1 = 0.f;
    {
        int c0 = col0 + lr, c1 = col0 + 16 + lr;
        if (c0 < J) bv0 = bias[c0];
        if (c1 < J) bv1 = bias[c1];
    }
#pragma unroll
    for (int ia = 0; ia < 2; ++ia) {
#pragma unroll
        for (int jb = 0; jb < 2; ++jb) {
            v8f acc = (ia == 0) ? (jb == 0 ? acc00 : acc01)
                                : (jb == 0 ? acc10 : acc11);
            float bval = jb == 0 ? bv0 : bv1;
            int colb = col0 + jb * 16 + lr;
#pragma unroll
            for (int r = 0; r < 8; ++r) {
                int row = row0 + ia * 16 + r + 8 * hi;
                if (row < rows && colb < J) {
                    size_t idx = (size_t)row * J + colb;
                    float v = acc[r] + bval;
                    if (ACCUM) v += C[idx];
                    if (ACT == 1) v = sigm(v);
                    C[idx] = v;
                }
            }
        }
    }
}

// ---------------------------------------------------------------------------
// x = concat(in_data [B,256], prev_reads.transpose(1,0,2) [B, NH*M])
// ---------------------------------------------------------------------------
__global__ void concat_x_kernel(const float* __restrict__ in_data,
                                const float* __restrict__ prev_reads,
                                float* __restrict__ x) {
    int idx = blockIdx.x * blockDim.x + threadIdx.x;
    if (idx >= B_ * (IN_ + NH_ * M_)) return;
    int b = idx / (IN_ + NH_ * M_);
    int k = idx % (IN_ + NH_ * M_);
    float v;
    if (k < IN_) v = in_data[(size_t)b * IN_ + k];
    else {
        int kk = k - IN_;
        v = prev_reads[((size_t)(kk >> 6) * B_ + b) * M_ + (kk & 63)];
    }
    x[idx] = v;
}

// xo = concat(h [B,512], reads [NH,B,64] -> [B, NH*M])
__global__ void concat_xo_kernel(const float* __restrict__ h,
                                 const float* __restrict__ reads,
                                 float* __restrict__ xo) {
    int idx = blockIdx.x * blockDim.x + threadIdx.x;
    if (idx >= B_ * (H_ + NH_ * M_)) return;
    int b = idx / (H_ + NH_ * M_);
    int k = idx % (H_ + NH_ * M_);
    float v;
    if (k < H_) v = h[(size_t)b * H_ + k];
    else {
        int kk = k - H_;
        v = reads[((size_t)(kk >> 6) * B_ + b) * M_ + (kk & 63)];
    }
    xo[idx] = v;
}

// ---------------------------------------------------------------------------
// LSTM nonlinearity: gates [B, 4H] (i|f|g|o) -> h [B, H]
// ---------------------------------------------------------------------------
__global__ void lstm_kernel(const float* __restrict__ gates,
                            const float* __restrict__ c0,
                            float* __restrict__ h) {
    int idx = blockIdx.x * blockDim.x + threadIdx.x;
    if (idx >= B_ * H_) return;
    int b = idx / H_, j = idx % H_;
    const float* g = gates + (size_t)b * 4 * H_;
    float ig = g[j], fg = g[H_ + j], gg = g[2 * H_ + j], og = g[3 * H_ + j];
    float c = sigm(fg) * c0[idx] + sigm(ig) * tanhf(gg);
    h[idx] = sigm(og) * tanhf(c);
}

// ---------------------------------------------------------------------------
// Head parameter projections: p[head][b][j] = h[b,:] . W[head][j,:] + bias
// ---------------------------------------------------------------------------
__global__ void proj_kernel(const float* __restrict__ h,
                            const float* __restrict__ W,
                            const float* __restrict__ bias,
                            float* __restrict__ outp, int J) {
    int head = blockIdx.y;
    int id = blockIdx.x * blockDim.x + threadIdx.x;
    if (id >= B_ * J) return;
    int b = id / J, j = id % J;
    const float* hr = h + (size_t)b * H_;
    const float* wr = W + ((size_t)head * J + j) * H_;
    float s = 0.f;
#pragma unroll 8
    for (int k = 0; k < H_; ++k) s += hr[k] * wr[k];
    outp[((size_t)head * B_ + b) * J + j] = s + bias[(size_t)head * J + j];
}

// ---------------------------------------------------------------------------
// Per (b, head, type): normalized key kn, scalar params, erase/add vectors.
// grid = (B, NH, 2), block = 64.
// ---------------------------------------------------------------------------
__global__ void prep_kernel(const float* __restrict__ p_r,
                            const float* __restrict__ p_w,
                            float* __restrict__ kn_r, float* __restrict__ kn_w,
                            float* __restrict__ ebuf, float* __restrict__ abuf,
                            float* __restrict__ params) {
    int b = blockIdx.x, head = blockIdx.y, type = blockIdx.z;
    int m = threadIdx.x;
    const float* p = (type == 0) ? p_r + ((size_t)head * B_ + b) * (M_ + 6)
                                 : p_w + ((size_t)head * B_ + b) * (3 * M_ + 6);
    float* kn = ((type == 0) ? kn_r : kn_w) + ((size_t)head * B_ + b) * M_;

    __shared__ float sd[64];
    float kv = tanhf(p[m]);
    sd[m] = kv * kv;
    __syncthreads();
    for (int s = 32; s > 0; s >>= 1) {
        if (m < s) sd[m] += sd[m + s];
        __syncthreads();
    }
    float nrm = sqrtf(sd[0]) + EPS_;
    kn[m] = kv / nrm;

    if (m == 0) {
        float beta  = softplus(p[M_]);
        float g     = sigm(p[M_ + 1]);
        float s0 = p[M_ + 2], s1 = p[M_ + 3], s2 = p[M_ + 4];
        float mx = fmaxf(s0, fmaxf(s1, s2));
        float e0 = expf(s0 - mx), e1 = expf(s1 - mx), e2 = expf(s2 - mx);
        float es = e0 + e1 + e2;
        float gamma = 1.f + softplus(p[M_ + 5]);
        float* prm = params + (((size_t)head * 2 + type) * B_ + b) * 6;
        prm[0] = beta; prm[1] = g;
        prm[2] = e0 / es; prm[3] = e1 / es; prm[4] = e2 / es;
        prm[5] = gamma;
    }
    if (type == 1) {
        ebuf[((size_t)head * B_ + b) * M_ + m] = sigm(p[M_ + 6 + m]);
        abuf[((size_t)head * B_ + b) * M_ + m] = tanhf(p[2 * M_ + 6 + m]);
    }
}

// ---------------------------------------------------------------------------
// Content scores: one wave32 per memory row (b,n).
// s = dot(row, kn) / (||row|| + eps)   for both read and write keys.
// ---------------------------------------------------------------------------
__global__ void score_kernel(const float* __restrict__ mem,
                             const float* __restrict__ kn_r,
                             const float* __restrict__ kn_w,
                             float* __restrict__ scores_r,
                             float* __restrict__ scores_w) {
    int lane = threadIdx.x & 31;
    int w = (blockIdx.x * blockDim.x + threadIdx.x) >> 5;
    if (w >= B_ * N_) return;
    int b = w / N_;
    const float* row = mem + (size_t)w * M_;
    float v0 = row[lane], v1 = row[lane + 32];
    const float* kr = kn_r + (size_t)b * M_;
    const float* kw = kn_w + (size_t)b * M_;
    float ss = v0 * v0 + v1 * v1;
    float dr = v0 * kr[lane] + v1 * kr[lane + 32];
    float dw = v0 * kw[lane] + v1 * kw[lane + 32];
#pragma unroll
    for (int msk = 16; msk >= 1; msk >>= 1) {
        ss += __shfl_xor(ss, msk);
        dr += __shfl_xor(dr, msk);
        dw += __shfl_xor(dw, msk);
    }
    if (lane == 0) {
        float inv = 1.f / (sqrtf(ss) + EPS_);
        scores_r[w] = dr * inv;
        scores_w[w] = dw * inv;
    }
}

// ---------------------------------------------------------------------------
// Addressing: softmax(beta*s) -> gate with prev -> circular shift -> sharpen.
// grid = (B, 2[type]), block = 256.  N=1024 lives in LDS for the roll.
// ---------------------------------------------------------------------------
__global__ void addr_kernel(const float* __restrict__ params,
                            const float* __restrict__ scores_r,
                            const float* __restrict__ scores_w,
                            const float* __restrict__ prev_r,
                            const float* __restrict__ prev_w,
                            float* __restrict__ w_r, float* __restrict__ w_w,
                            int head) {
    int b = blockIdx.x, type = blockIdx.y;
    int t = threadIdx.x;
    const float* sc = ((type == 0) ? scores_r : scores_w) + (size_t)b * N_;
    const float* pv = ((type == 0) ? prev_r   : prev_w)   + (size_t)b * N_;
    float* outw     = ((type == 0) ? w_r      : w_w)      + (size_t)b * N_;
    const float* prm = params + (((size_t)head * 2 + type) * B_ + b) * 6;
    float beta = prm[0], g = prm[1];
    float sh0 = prm[2], sh1 = prm[3], sh2 = prm[4], gamma = prm[5];

    __shared__ float arr[N_];
    __shared__ float red[256];

    float lmax = -3.402823466e38f;
    for (int n = t; n < N_; n += 256) {
        float v = beta * sc[n];
        arr[n] = v;
        lmax = fmaxf(lmax, v);
    }
    red[t] = lmax; __syncthreads();
    for (int s = 128; s > 0; s >>= 1) {
        if (t < s) red[t] = fmaxf(red[t], red[t + s]);
        __syncthreads();
    }
    float mx = red[0]; __syncthreads();

    float lsum = 0.f;
    for (int n = t; n < N_; n += 256) {
        float e = expf(arr[n] - mx);
        arr[n] = e;
        lsum += e;
    }
    red[t] = lsum; __syncthreads();
    for (int s = 128; s > 0; s >>= 1) {
        if (t < s) red[t] += red[t + s];
        __syncthreads();
    }
    float ssum = red[0]; __syncthreads();

    for (int n = t; n < N_; n += 256) {
        float wc = arr[n] / ssum;
        arr[n] = g * wc + (1.f - g) * pv[n];
    }
    __syncthreads();

    float wp[4];
    float lps = 0.f;
#pragma unroll
    for (int i = 0; i < 4; ++i) {
        int n = t + i * 256;
        float ws = sh0 * arr[(n + N_ - 1) & (N_ - 1)]
                 + sh1 * arr[n]
                 + sh2 * arr[(n + 1) & (N_ - 1)];
        wp[i] = powf(ws, gamma);
        lps += wp[i];
    }
    red[t] = lps; __syncthreads();
    for (int s = 128; s > 0; s >>= 1) {
        if (t < s) red[t] += red[t + s];
        __syncthreads();
    }
    float inv = 1.f / (red[0] + EPS_);
#pragma unroll
    for (int i = 0; i < 4; ++i) outw[t + i * 256] = wp[i] * inv;
}

// ---------------------------------------------------------------------------
// Fused read + erase/add write. grid = B, block = 256 (4 n-rows in flight).
// Read uses pre-write memory (matches reference ordering).
// ---------------------------------------------------------------------------
__global__ void rw_kernel(float* __restrict__ mem,
                          const float* __restrict__ w_r,
                          const float* __restrict__ w_w,
                          const float* __restrict__ ebuf,
                          const float* __restrict__ abuf,
                          float* __restrict__ readout) {
    int b = blockIdx.x;
    int t = threadIdx.x, ty = t >> 6, m = t & 63;
    __shared__ float part[4][64];
    const float* wr = w_r + (size_t)b * N_;
    const float* ww = w_w + (size_t)b * N_;
    float ev = ebuf[(size_t)b * M_ + m];
    float av = abuf[(size_t)b * M_ + m];
    float* mb = mem + (size_t)b * N_ * M_;
    float acc = 0.f;
    for (int n = ty; n < N_; n += 4) {
        float wrv = wr[n], wwv = ww[n];
        float v = mb[(size_t)n * M_ + m];
        acc += wrv * v;
        mb[(size_t)n * M_ + m] = v * (1.f - wwv * ev) + wwv * av;
    }
    part[ty][m] = acc;
    __syncthreads();
    if (ty == 0)
        readout[(size_t)b * M_ + m] = part[0][m] + part[1][m] + part[2][m] + part[3][m];
}

// ---------------------------------------------------------------------------
extern "C" void kernel_launch(void* const* d_in, const int* in_sizes, int n_in,
                              void* d_out, int out_size, void* d_ws, size_t ws_size,
                              hipStream_t stream) {
    (void)in_sizes; (void)n_in; (void)out_size; (void)ws_size;
    const float* in_data    = (const float*)d_in[0];
    const float* memory     = (const float*)d_in[1];
    const float* prev_reads = (const float*)d_in[2];
    const float* prev_w_r   = (const float*)d_in[3];
    const float* prev_w_w   = (const float*)d_in[4];
    const float* h0         = (const float*)d_in[5];
    const float* c0         = (const float*)d_in[6];
    const float* W_ih       = (const float*)d_in[7];
    const float* b_ih       = (const float*)d_in[8];
    const float* W_hh       = (const float*)d_in[9];
    const float* b_hh       = (const float*)d_in[10];
    const float* W_rh       = (const float*)d_in[11];
    const float* b_rh       = (const float*)d_in[12];
    const float* W_wh       = (const float*)d_in[13];
    const float* b_wh       = (const float*)d_in[14];
    const float* W_out      = (const float*)d_in[15];
    const float* b_out      = (const float*)d_in[16];
    float* out = (float*)d_out;

    char* base = (char*)d_ws;
    size_t off = 0;
    auto take = [&](size_t elems) -> float* {
        float* p = (float*)(base + off);
        off = (off + elems * sizeof(float) + 255) & ~(size_t)255;
        return p;
    };
    float* memws    = take((size_t)B_ * N_ * M_);
    float* xbuf     = take((size_t)B_ * (IN_ + NH_ * M_));
    float* gates    = take((size_t)B_ * 4 * H_);
    float* hbuf     = take((size_t)B_ * H_);
    float* p_r      = take((size_t)NH_ * B_ * (M_ + 6));
    float* p_w      = take((size_t)NH_ * B_ * (3 * M_ + 6));
    float* kn_r     = take((size_t)NH_ * B_ * M_);
    float* kn_w     = take((size_t)NH_ * B_ * M_);
    float* ebuf     = take((size_t)NH_ * B_ * M_);
    float* abuf     = take((size_t)NH_ * B_ * M_);
    float* params   = take((size_t)NH_ * 2 * B_ * 6);
    float* scores_r = take((size_t)B_ * N_);
    float* scores_w = take((size_t)B_ * N_);
    float* wrbuf    = take((size_t)B_ * N_);
    float* wwbuf    = take((size_t)B_ * N_);
    float* reads    = take((size_t)NH_ * B_ * M_);
    float* xo       = take((size_t)B_ * (H_ + NH_ * M_));

    // Working copy of memory (inputs must not be mutated).
    hipMemcpyAsync(memws, memory, (size_t)B_ * N_ * M_ * sizeof(float),
                   hipMemcpyDeviceToDevice, stream);

    // 1) x = [in_data | prev_reads]
    {
        int n = B_ * (IN_ + NH_ * M_);
        concat_x_kernel<<<(n + 255) / 256, 256, 0, stream>>>(in_data, prev_reads, xbuf);
    }
    // 2) gates = x @ W_ih^T + b_ih ; gates += h0 @ W_hh^T + b_hh  (WMMA f32)
    {
        int tiles = (B_ / 32) * (4 * H_ / 32);
        int blocks = (tiles + 3) / 4;
        wmma_gemm_f32<0, 0><<<blocks, 128, 0, stream>>>(
            xbuf, W_ih, b_ih, gates, B_, IN_ + NH_ * M_, 4 * H_);
        wmma_gemm_f32<0, 1><<<blocks, 128, 0, stream>>>(
            h0, W_hh, b_hh, gates, B_, H_, 4 * H_);
    }
    // 3) LSTM -> h
    lstm_kernel<<<(B_ * H_ + 255) / 256, 256, 0, stream>>>(gates, c0, hbuf);

    // 4) head projections p_r [NH,B,70], p_w [NH,B,198]
    {
        int Jr = M_ + 6, Jw = 3 * M_ + 6;
        dim3 gr((B_ * Jr + 255) / 256, NH_), gw((B_ * Jw + 255) / 256, NH_);
        proj_kernel<<<gr, 256, 0, stream>>>(hbuf, W_rh, b_rh, p_r, Jr);
        proj_kernel<<<gw, 256, 0, stream>>>(hbuf, W_wh, b_wh, p_w, Jw);
    }
    // 5) keys / scalars / erase-add vectors
    {
        dim3 g(B_, NH_, 2);
        prep_kernel<<<g, 64, 0, stream>>>(p_r, p_w, kn_r, kn_w, ebuf, abuf, params);
    }
    // 6) sequential heads (head i sees writes of heads < i)
    for (int hd = 0; hd < NH_; ++hd) {
        int waves = B_ * N_;
        score_kernel<<<waves / 8, 256, 0, stream>>>(
            memws, kn_r + (size_t)hd * B_ * M_, kn_w + (size_t)hd * B_ * M_,
            scores_r, scores_w);
        dim3 ga(B_, 2);
        addr_kernel<<<ga, 256, 0, stream>>>(
            params, scores_r, scores_w,
            prev_w_r + (size_t)hd * B_ * N_, prev_w_w + (size_t)hd * B_ * N_,
            wrbuf, wwbuf, hd);
        rw_kernel<<<B_, 256, 0, stream>>>(
            memws, wrbuf, wwbuf,
            ebuf + (size_t)hd * B_ * M_, abuf + (size_t)hd * B_ * M_,
            reads + (size_t)hd * B_ * M_);
    }
    // 7) out = sigmoid([h | reads] @ W_out^T + b_out)
    {
        int n = B_ * (H_ + NH_ * M_);
        concat_xo_kernel<<<(n + 255) / 256, 256, 0, stream>>>(hbuf, reads, xo);
        int tiles = (B_ / 32) * (256 / 32);
        int blocks = (tiles + 3) / 4;
        wmma_gemm_f32<1, 0><<<blocks, 128, 0, stream>>>(
            xo, W_out, b_out, out, B_, H_ + NH_ * M_, 256);
    }
}